// DAAutoencoder_90443421319698
// MI455X (gfx1250) — compile-verified
//
#include <hip/hip_runtime.h>
#include <hip/hip_bf16.h>
#include <stdint.h>

// ---------------- CDNA5 async Global->LDS path (guarded) ----------------
#ifndef __HIP_DEVICE_COMPILE__
#define DAA_HAVE_ASYNC 0
#else
#if __has_builtin(__builtin_amdgcn_global_load_async_to_lds_b32) && \
    __has_builtin(__builtin_amdgcn_s_wait_asynccnt)
#define DAA_HAVE_ASYNC 1
#else
#define DAA_HAVE_ASYNC 0
#endif
#endif

#if DAA_HAVE_ASYNC
typedef __attribute__((address_space(1))) int glb_int_t;  // global (device) AS
typedef __attribute__((address_space(3))) int lds_int_t;  // LDS (shared) AS
// async DMA: GLOBAL_LOAD_ASYNC_TO_LDS_B32, tracked by ASYNCcnt
#define ASYNC_CP_B32(gp, lp) \
  __builtin_amdgcn_global_load_async_to_lds_b32((glb_int_t*)(gp), (lds_int_t*)(lp), 0, 0)
#define WAIT_ASYNC(n) __builtin_amdgcn_s_wait_asynccnt(n)
#else
#define ASYNC_CP_B32(gp, lp) (*(lp) = *(gp))
#define WAIT_ASYNC(n) ((void)0)
#endif

__device__ __forceinline__ unsigned ballot32(bool pred) {
#if defined(__HIP_DEVICE_COMPILE__) && __has_builtin(__builtin_amdgcn_ballot_w32)
  return __builtin_amdgcn_ballot_w32(pred);            // wave32 ballot
#else
  return (unsigned)__ballot(pred);
#endif
}

#define TPB 128          // threads per block = batch rows per block (4 wave32s)
#define CHUNK 32         // input features per LDS tile chunk (== 1 mask word)
#define TILE_STRIDE 33   // +1 pad -> stride 33 floats: conflict-free (gcd(33,64)=1)

// One DAA layer: y[b, node_base+n] = reduce_{i : sel[n,i]==1} x[b,i], identity = OFF.
// Block: 128 threads = 128 batch rows; NPB output nodes per block.
template <int IN, int OUT, int NPB, bool IS_MIN>
__global__ __launch_bounds__(TPB) void daa_layer_kernel(
    const float* __restrict__ x, const int* __restrict__ sel,
    float* __restrict__ y) {
  constexpr int NCHUNK = IN / CHUNK;
  constexpr int WPN = IN / 32;                 // mask words per node
  constexpr float OFF = IS_MIN ? 2.0f : -1.0f; // pushes no_edge out of range

  __shared__ float tile[2][TPB][TILE_STRIDE];  // x chunk, [row][feature], padded
  __shared__ unsigned maskbuf[NPB * WPN];      // 1 bit per (node, in-feature)

  const int tid = (int)threadIdx.x;
  const int lane = tid & 31;
  const int wave = tid >> 5;                   // 0..3
  const int node_base = (int)blockIdx.x * NPB;
  const int bbase = (int)blockIdx.y * TPB;

  // ---- pack selection bits once per block via wave32 ballot ----
  for (int w = wave; w < NPB * WPN; w += (TPB / 32)) {
    const int n = w / WPN;
    const int iw = w - n * WPN;
    const int s = sel[(size_t)(node_base + n) * IN + (size_t)iw * 32 + lane];
    const unsigned m = ballot32(s == 1);
    if (lane == 0) maskbuf[w] = m;
  }

  // ---- double-buffered async Global->LDS staging of x tiles ----
  // Wave w issues rows r == w (mod 4); lanes cover 32 consecutive features
  // (128B coalesced global read, contiguous 128B LDS write per instruction).
  auto issue = [&](int c, int buf) {
#pragma unroll
    for (int p = 0; p < CHUNK; ++p) {
      const int r = p * (TPB / 32) + wave;     // 0..127
      const float* gp = x + (size_t)(bbase + r) * IN + (size_t)c * CHUNK + lane;
      float* lp = &tile[buf][r][lane];
      ASYNC_CP_B32(gp, lp);
    }
  };

  issue(0, 0);
  if (NCHUNK > 1) issue(1, 1);

  float acc[NPB];
#pragma unroll
  for (int n = 0; n < NPB; ++n) acc[n] = OFF;

  for (int c = 0; c < NCHUNK; ++c) {
    // Drain own ASYNCcnt down to the ops of the prefetched chunk, then make
    // all waves' tiles visible with a workgroup barrier.
    if (c + 1 < NCHUNK) { WAIT_ASYNC(CHUNK); } else { WAIT_ASYNC(0); }
    __syncthreads();

    const int buf = c & 1;
    unsigned m[NPB];
#pragma unroll
    for (int n = 0; n < NPB; ++n) m[n] = maskbuf[n * WPN + c];  // uniform LDS broadcast

#pragma unroll
    for (int i = 0; i < CHUNK; ++i) {
      const float xv = tile[buf][tid][i];      // stride-33 rows: conflict-free
#pragma unroll
      for (int n = 0; n < NPB; ++n) {
        const bool e = (m[n] >> i) & 1u;
        const float cand = IS_MIN ? fminf(acc[n], xv) : fmaxf(acc[n], xv);
        acc[n] = e ? cand : acc[n];            // branchless select+reduce
      }
    }

    __syncthreads();                            // tile fully consumed
    if (c + 2 < NCHUNK) issue(c + 2, buf);      // overlap DMA with next compute
  }

  // y is [B, OUT] row-major; NPB consecutive floats per thread (vectorizable).
  float* yp = y + (size_t)(bbase + tid) * OUT + node_base;
#pragma unroll
  for (int n = 0; n < NPB; ++n) yp[n] = acc[n];
}

extern "C" void kernel_launch(void* const* d_in, const int* in_sizes, int n_in,
                              void* d_out, int out_size, void* d_ws, size_t ws_size,
                              hipStream_t stream) {
  (void)in_sizes; (void)n_in; (void)out_size; (void)ws_size;

  constexpr int B = 256, IN = 1024, HID = 512;

  const float* x   = (const float*)d_in[0];
  const int*  sel0 = (const int*)d_in[1];   // [HID, IN]
  const int*  sel1 = (const int*)d_in[2];   // [IN, HID]
  float* out = (float*)d_out;               // [B, IN]
  float* h   = (float*)d_ws;                // [B, HID] intermediate (512 KB)

  // Layer 0: T-norm (min), 4 nodes/block -> grid (128, 2)
  dim3 g0(HID / 4, B / TPB);
  daa_layer_kernel<IN, HID, 4, true><<<g0, TPB, 0, stream>>>(x, sel0, h);

  // Layer 1: T-conorm (max), 8 nodes/block -> grid (128, 2)
  dim3 g1(IN / 8, B / TPB);
  daa_layer_kernel<HID, IN, 8, false><<<g1, TPB, 0, stream>>>(h, sel1, out);
}